// SlidingWindowAttention_36953898615483
// MI455X (gfx1250) — compile-verified
//
#include <hip/hip_runtime.h>

typedef float v2f __attribute__((ext_vector_type(2)));
typedef float v8f __attribute__((ext_vector_type(8)));

#if __has_builtin(__builtin_amdgcn_sched_barrier)
#define SCHED_BARRIER() __builtin_amdgcn_sched_barrier(0)
#else
#define SCHED_BARRIER() asm volatile("" ::: "memory")
#endif

namespace {
constexpr int   kBatch = 2;
constexpr int   kSeq   = 2048;
constexpr int   kDim   = 128;     // embed dim
constexpr int   kWin   = 128;     // keys in [q - 128, q]  (129 positions)
constexpr float kNegInf = -1.0e9f;
constexpr float kScale  = 0.08838834764831845f;   // 1/sqrt(128)
constexpr int   kTilesPerSeq   = kSeq / 16;       // 128
constexpr int   kWavesPerBlock = 4;
}

// One wave32 handles one 16-query tile. Scores computed transposed
// (PT[n][m] = k_n . q_m) so the softmax reduction over keys is
// per-lane over the 8 D-layout VGPRs + one shfl_xor(16) across halves.
__global__ __launch_bounds__(128)
void swa_wmma_f32_kernel(const float* __restrict__ Q,
                         const float* __restrict__ K,
                         const float* __restrict__ V,
                         float* __restrict__ O) {
  const int lane = threadIdx.x & 31;
  const int wave = threadIdx.x >> 5;
  const int tile = blockIdx.x * kWavesPerBlock + wave;   // 0..255
  const int b    = tile / kTilesPerSeq;
  const int q0   = (tile % kTilesPerSeq) * 16;

  const int h   = lane >> 4;     // lane half (0/1)
  const int l15 = lane & 15;

  const float* Qb = Q + (size_t)b * kSeq * kDim;
  const float* Kb = K + (size_t)b * kSeq * kDim;
  const float* Vb = V + (size_t)b * kSeq * kDim;
  float*       Ob = O + (size_t)b * kSeq * kDim;

  // Resident Q tile as B-fragments of the K x Q^T GEMM.
  // chunk c covers embed dims 4c..4c+3:
  //   v0(lane) = Q[q0+l15][4c+2h], v1(lane) = Q[q0+l15][4c+2h+1]  -> one float2 load
  v2f qb[32];
  {
    const float* qrow = Qb + (size_t)(q0 + l15) * kDim + 2 * h;
#pragma unroll
    for (int c = 0; c < 32; ++c)
      qb[c] = *(const v2f*)(qrow + 4 * c);
  }

  // Output accumulators: O^T[d][m], 8 chunks of 16 dims (D-layout).
  v8f acc[8];
#pragma unroll
  for (int i = 0; i < 8; ++i) acc[i] = (v8f)0.0f;

  float run_max = kNegInf;
  float run_sum = 0.0f;

  // 9 key tiles cover keys q0-128 .. q0+15
#pragma unroll 1
  for (int j = 0; j < 9; ++j) {
    const int kb = q0 - kWin + 16 * j;

    // ---- stage 1: load ALL 32 K-fragments into distinct registers first.
    v2f ka[32];
    {
      int krow = kb + l15;
      if (krow < 0) krow = 0;                    // clamp; masked below
      const float* kptr = Kb + (size_t)krow * kDim + 2 * h;
#pragma unroll
      for (int c = 0; c < 32; ++c)
        ka[c] = *(const v2f*)(kptr + 4 * c);
    }
    // Forbid the scheduler from sinking the loads into the WMMA chain:
    // all 32 stay outstanding, waits become progressive.
    SCHED_BARRIER();

    // ---- stage 2: scores^T tile PT[n][m] = k_{kb+n} . q_{q0+m}
    // Two independent accumulation chains (even/odd chunks) for WMMA ILP.
    v8f pt0 = (v8f)0.0f, pt1 = (v8f)0.0f;
#pragma unroll
    for (int c = 0; c < 16; ++c) {
      pt0 = __builtin_amdgcn_wmma_f32_16x16x4_f32(false, ka[2 * c],     false, qb[2 * c],
                                                  (short)0, pt0, false, false);
      pt1 = __builtin_amdgcn_wmma_f32_16x16x4_f32(false, ka[2 * c + 1], false, qb[2 * c + 1],
                                                  (short)0, pt1, false, false);
    }
    v8f pt;
#pragma unroll
    for (int r = 0; r < 8; ++r) pt[r] = pt0[r] + pt1[r];

    // ---- stage 3: preload V fragments for d-chunks 0..3 (independent of the
    // softmax; free to co-schedule with the WMMA chain above).
    // A-frag for chunk c: v0 = V[kb+4c+2h][dbase+l15], v1 = row+1.
    int vrow0[4], vrow1[4];
#pragma unroll
    for (int c = 0; c < 4; ++c) {
      int r0 = kb + 4 * c + 2 * h;
      int r1 = r0 + 1;
      vrow0[c] = (r0 < 0) ? 0 : r0;              // clamp; weights are 0 when masked
      vrow1[c] = (r1 < 0) ? 0 : r1;
    }
    v2f va0[16];                                  // [dc*4 + c], dc = 0..3
#pragma unroll
    for (int dc = 0; dc < 4; ++dc) {
      const float* vcol = Vb + dc * 16 + l15;
#pragma unroll
      for (int c = 0; c < 4; ++c) {
        va0[dc * 4 + c][0] = vcol[(size_t)vrow0[c] * kDim];
        va0[dc * 4 + c][1] = vcol[(size_t)vrow1[c] * kDim];
      }
    }
    // Keep the va0 batch above the softmax (don't let it sink to its use).
    SCHED_BARRIER();

    // ---- stage 4: mask + scale; PT reg r, this lane: key n = r + 8h, query m = l15
    float pmax = kNegInf;
#pragma unroll
    for (int r = 0; r < 8; ++r) {
      const int key  = kb + r + 8 * h;
      const int qidx = q0 + l15;
      const bool valid = (key >= 0) && (key <= qidx) && (key >= qidx - kWin);
      const float s = valid ? pt[r] * kScale : kNegInf;
      pt[r] = s;
      pmax = fmaxf(pmax, s);
    }
    pmax = fmaxf(pmax, __shfl_xor(pmax, 16, 32));   // combine lane halves

    const float new_max = fmaxf(run_max, pmax);
    const float corr    = __expf(run_max - new_max); // underflows to 0 past first valid tile
    run_max = new_max;

    float psum = 0.0f;
#pragma unroll
    for (int r = 0; r < 8; ++r) {
      const float e = __expf(pt[r] - new_max);
      pt[r] = e;
      psum += e;
    }
    psum += __shfl_xor(psum, 16, 32);
    run_sum = run_sum * corr + psum;

#pragma unroll
    for (int i = 0; i < 8; ++i)
#pragma unroll
      for (int r = 0; r < 8; ++r)
        acc[i][r] *= corr;

    // ---- stage 5: repack PT (D-layout) into B-fragments for O^T += V^T x PT
    // chunk c needs rows n = 4c..4c+3 of PT; row n lives in reg (n&7), half (n>>3).
    v2f pb[4];
    {
      const float x2 = __shfl_xor(pt[2], 16, 32);
      const float x3 = __shfl_xor(pt[3], 16, 32);
      pb[0][0] = h ? x2 : pt[0];  pb[0][1] = h ? x3 : pt[1];
      const float x6 = __shfl_xor(pt[6], 16, 32);
      const float x7 = __shfl_xor(pt[7], 16, 32);
      pb[1][0] = h ? x6 : pt[4];  pb[1][1] = h ? x7 : pt[5];
      const float y0 = __shfl_xor(pt[0], 16, 32);
      const float y1 = __shfl_xor(pt[1], 16, 32);
      pb[2][0] = h ? pt[2] : y0;  pb[2][1] = h ? pt[3] : y1;
      const float y4 = __shfl_xor(pt[4], 16, 32);
      const float y5 = __shfl_xor(pt[5], 16, 32);
      pb[3][0] = h ? pt[6] : y4;  pb[3][1] = h ? pt[7] : y5;
    }

    // ---- stage 6: O^T accumulation, first half (d-chunks 0..3).
    // 4 independent accumulator chains -> plenty of WMMA ILP.
#pragma unroll
    for (int dc = 0; dc < 4; ++dc)
#pragma unroll
      for (int c = 0; c < 4; ++c)
        acc[dc] = __builtin_amdgcn_wmma_f32_16x16x4_f32(false, va0[dc * 4 + c],
                                                        false, pb[c],
                                                        (short)0, acc[dc], false, false);

    // ---- stage 7: load V fragments for d-chunks 4..7, then accumulate
    v2f va1[16];
#pragma unroll
    for (int dc = 0; dc < 4; ++dc) {
      const float* vcol = Vb + (dc + 4) * 16 + l15;
#pragma unroll
      for (int c = 0; c < 4; ++c) {
        va1[dc * 4 + c][0] = vcol[(size_t)vrow0[c] * kDim];
        va1[dc * 4 + c][1] = vcol[(size_t)vrow1[c] * kDim];
      }
    }
    SCHED_BARRIER();
#pragma unroll
    for (int dc = 0; dc < 4; ++dc)
#pragma unroll
      for (int c = 0; c < 4; ++c)
        acc[dc + 4] = __builtin_amdgcn_wmma_f32_16x16x4_f32(false, va1[dc * 4 + c],
                                                            false, pb[c],
                                                            (short)0, acc[dc + 4], false, false);
  }

  // ---- normalize and store: acc[dc] reg r -> d = dc*16 + r + 8h, query = q0 + l15
  const float inv = 1.0f / run_sum;
  float* orow = Ob + (size_t)(q0 + l15) * kDim;
#pragma unroll
  for (int dc = 0; dc < 8; ++dc)
#pragma unroll
    for (int r = 0; r < 8; ++r)
      orow[dc * 16 + r + 8 * h] = acc[dc][r] * inv;
}

extern "C" void kernel_launch(void* const* d_in, const int* in_sizes, int n_in,
                              void* d_out, int out_size, void* d_ws, size_t ws_size,
                              hipStream_t stream) {
  const float* q = (const float*)d_in[0];
  const float* k = (const float*)d_in[1];
  const float* v = (const float*)d_in[2];
  float* out = (float*)d_out;

  const int total_tiles = kBatch * kTilesPerSeq;          // 256
  dim3 grid(total_tiles / kWavesPerBlock);                // 64
  dim3 block(32 * kWavesPerBlock);                        // 128 threads = 4 waves
  swa_wmma_f32_kernel<<<grid, block, 0, stream>>>(q, k, v, out);
}